// SplineInterpolate2D_39393440039347
// MI455X (gfx1250) — compile-verified
//
#include <hip/hip_runtime.h>
#include <math.h>
#include <stdint.h>

typedef float v2f __attribute__((ext_vector_type(2)));
typedef float v8f __attribute__((ext_vector_type(8)));

#define NXF   512            // fit grid size == number of basis functions
#define NEV   1024           // eval grid size
#define KORD  3              // cubic
#define NKNOT (NXF + KORD + 1)   // 516
#define BC    64             // BATCH*CH = 16*4

#define KC    16             // K-panel staged in LDS
#define LDB   136            // LDS row pitch (floats): 16B-aligned, conflict-free

// ---------------------------------------------------------------------------
// B-spline helpers (device)
// ---------------------------------------------------------------------------
__device__ __forceinline__ int find_span(const float* __restrict__ t, float x) {
  if (x >= t[NXF])  return NXF - 1;
  if (x <= t[KORD]) return KORD;
  int lo = KORD, hi = NXF;
  while (hi - lo > 1) { int mid = (lo + hi) >> 1; if (t[mid] <= x) lo = mid; else hi = mid; }
  return lo;
}

__device__ __forceinline__ void basis_funs(const float* __restrict__ t, int span,
                                           float x, float* __restrict__ Nv) {
  float L[KORD + 1], R[KORD + 1];
  Nv[0] = 1.f;
  for (int j = 1; j <= KORD; ++j) {
    L[j] = x - t[span + 1 - j];
    R[j] = t[span + j] - x;
    float saved = 0.f;
    for (int r = 0; r < j; ++r) {
      float denom = R[r + 1] + L[j - r];
      float temp  = (denom != 0.f) ? Nv[r] / denom : 0.f;
      Nv[r] = saved + R[r + 1] * temp;
      saved = L[j - r] * temp;
    }
    Nv[j] = saved;
  }
}

// ---------------------------------------------------------------------------
// Setup kernels
// ---------------------------------------------------------------------------
__global__ void build_knots(float* __restrict__ t) {
  int j = blockIdx.x * blockDim.x + threadIdx.x;
  if (j >= NKNOT) return;
  float v;
  if (j <= KORD)      v = -1.f;
  else if (j >= NXF)  v =  1.f;
  else                v = -1.f + 2.f * (float)(j - 2) / 511.f;
  t[j] = v;
}

__global__ void build_basis(const float* __restrict__ t, float* __restrict__ Bout, int P) {
  int p = blockIdx.x;
  float x = -1.f + 2.f * (float)p / (float)(P - 1);
  x = fminf(fmaxf(x, t[0]), t[NKNOT - 1]);
  int span = find_span(t, x);
  float Nv[KORD + 1];
  basis_funs(t, span, x, Nv);
  int c0 = span - KORD;
  for (int c = threadIdx.x; c < NXF; c += blockDim.x) {
    int r = c - c0;
    float v = (r >= 0 && r <= KORD) ? Nv[r] : 0.f;
    Bout[(long long)p * NXF + c] = v;
  }
}

__global__ void transpose512(const float* __restrict__ A, float* __restrict__ At) {
  int idx = blockIdx.x * blockDim.x + threadIdx.x;   // < 512*512
  int r = idx >> 9, c = idx & 511;
  At[(long long)c * NXF + r] = A[idx];
}

__global__ void gram_band(const float* __restrict__ Bmat, float* __restrict__ Gb) {
  int j = blockIdx.x * blockDim.x + threadIdx.x;
  if (j >= NXF) return;
  float s0 = 0.f, s1 = 0.f, s2 = 0.f, s3 = 0.f;
  for (int i = 0; i < NXF; ++i) {
    float bj = Bmat[(long long)i * NXF + j];
    s0 += bj * bj;
    if (j + 1 < NXF) s1 += bj * Bmat[(long long)i * NXF + j + 1];
    if (j + 2 < NXF) s2 += bj * Bmat[(long long)i * NXF + j + 2];
    if (j + 3 < NXF) s3 += bj * Bmat[(long long)i * NXF + j + 3];
  }
  Gb[j * 4 + 0] = s0; Gb[j * 4 + 1] = s1; Gb[j * 4 + 2] = s2; Gb[j * 4 + 3] = s3;
}

__global__ void chol_band(const float* __restrict__ Gb, float* __restrict__ Lo) {
  if (threadIdx.x != 0 || blockIdx.x != 0) return;
  for (int j = 0; j < NXF; ++j) {
    for (int d = 3; d >= 1; --d) {
      float val = 0.f;
      if (d <= j) {
        int i = j - d;
        float s = Gb[i * 4 + d];
        for (int m = j - 3; m < i; ++m) {
          if (m < 0) continue;
          s -= Lo[j * 4 + (j - m)] * Lo[i * 4 + (i - m)];
        }
        val = s / Lo[i * 4];
      }
      Lo[j * 4 + d] = val;
    }
    float s = Gb[j * 4];
    for (int d = 1; d <= 3; ++d) { if (d <= j) s -= Lo[j * 4 + d] * Lo[j * 4 + d]; }
    Lo[j * 4] = sqrtf(fmaxf(s, 1e-30f));
  }
}

// ---------------------------------------------------------------------------
// Banded SPD solves (in-place), bandwidth 3.
// ---------------------------------------------------------------------------
__global__ __launch_bounds__(256)
void band_solve_cols(float* __restrict__ Tm, const float* __restrict__ Lo) {
  int idx = blockIdx.x * blockDim.x + threadIdx.x;
  int bc = idx >> 9, n = idx & 511;
  float* P = Tm + (long long)bc * (NXF * NXF) + n;
  float ym1 = 0.f, ym2 = 0.f, ym3 = 0.f;
  for (int j = 0; j < NXF; ++j) {
    const float* l = Lo + j * 4;
    float v = P[(long long)j * NXF] - l[1] * ym1 - l[2] * ym2 - l[3] * ym3;
    v /= l[0];
    P[(long long)j * NXF] = v;
    ym3 = ym2; ym2 = ym1; ym1 = v;
  }
  float x1 = 0.f, x2 = 0.f, x3 = 0.f;
  for (int j = NXF - 1; j >= 0; --j) {
    float v = P[(long long)j * NXF];
    if (j + 1 < NXF) v -= Lo[(j + 1) * 4 + 1] * x1;
    if (j + 2 < NXF) v -= Lo[(j + 2) * 4 + 2] * x2;
    if (j + 3 < NXF) v -= Lo[(j + 3) * 4 + 3] * x3;
    v /= Lo[j * 4];
    P[(long long)j * NXF] = v;
    x3 = x2; x2 = x1; x1 = v;
  }
}

__global__ __launch_bounds__(256)
void band_solve_rows(float* __restrict__ Tm, const float* __restrict__ Lo) {
  int idx = blockIdx.x * blockDim.x + threadIdx.x;
  int bc = idx >> 9, j = idx & 511;
  float* P = Tm + (long long)bc * (NXF * NXF) + (long long)j * NXF;
  float ym1 = 0.f, ym2 = 0.f, ym3 = 0.f;
  for (int m = 0; m < NXF; ++m) {
    const float* l = Lo + m * 4;
    float v = P[m] - l[1] * ym1 - l[2] * ym2 - l[3] * ym3;
    v /= l[0];
    P[m] = v;
    ym3 = ym2; ym2 = ym1; ym1 = v;
  }
  float x1 = 0.f, x2 = 0.f, x3 = 0.f;
  for (int m = NXF - 1; m >= 0; --m) {
    float v = P[m];
    if (m + 1 < NXF) v -= Lo[(m + 1) * 4 + 1] * x1;
    if (m + 2 < NXF) v -= Lo[(m + 2) * 4 + 2] * x2;
    if (m + 3 < NXF) v -= Lo[(m + 3) * 4 + 3] * x3;
    v /= Lo[m * 4];
    P[m] = v;
    x3 = x2; x2 = x1; x1 = v;
  }
}

// ---------------------------------------------------------------------------
// FP32 WMMA GEMM, 2x2 register-blocked waves, double-buffered async
// global->LDS B staging (gfx1250 GLOBAL_LOAD_ASYNC_TO_LDS_B128 + ASYNCcnt).
//   C[M,N] = A[M,K] * op(B);  TRANSB: B stored [N,K] (direct K-contig loads).
//   Block: 8 waves in 2(M) x 4(N) grid; wave tile 32x32; block tile 64x128.
// Fragment layouts per CDNA5 ISA 7.12.2 (f32 16x16x4):
//   A: a[v] = A[m + lane%16][k + v + 2*(lane/16)]
//   B: b[v] = B[k + v + 2*(lane/16)][n + lane%16]
//   D: d[v] = C[m + v + 8*(lane/16)][n + lane%16]
// ---------------------------------------------------------------------------
__device__ __forceinline__ v8f wmma4(v2f a, v2f b, v8f c) {
  return __builtin_amdgcn_wmma_f32_16x16x4_f32(false, a, false, b, (short)0, c,
                                               false, false);
}

__device__ __forceinline__ void async_b128(uint32_t lds_off, const float* g) {
  asm volatile("global_load_async_to_lds_b128 %0, %1, off"
               :: "v"(lds_off), "v"(g) : "memory");
}

template <int TRANSB>
__global__ __launch_bounds__(256)
void gemm_wmma(const float* __restrict__ A, long long aB,
               const float* __restrict__ Bm, long long bB,
               float* __restrict__ C, long long cB,
               int M, int N, int K) {
  __shared__ float bs[2 * KC * LDB];

  const int lane = threadIdx.x;          // 0..31 (wave32)
  const int wave = threadIdx.y;          // 0..7
  const int row  = lane & 15;
  const int half = lane >> 4;
  const int wm   = wave >> 2;            // 0..1
  const int wn   = wave & 3;             // 0..3
  const int mW   = blockIdx.y * 64 + wm * 32;   // wave M origin
  const int nB   = blockIdx.x * 128;            // block N origin
  const int nW   = wn * 32;                     // wave N offset in block
  const int bc   = blockIdx.z;

  const float* Ab = A + (long long)bc * aB;
  const float* Bb = Bm + (long long)bc * bB;
  float* Cb = C + (long long)bc * cB;

  const float* a0 = Ab + (long long)(mW + row) * K + 2 * half;
  const float* a1 = a0 + 16LL * K;

  v8f acc00 = {0.f,0.f,0.f,0.f,0.f,0.f,0.f,0.f};
  v8f acc01 = acc00, acc10 = acc00, acc11 = acc00;

  if (TRANSB) {
    const float* b0 = Bb + (long long)(nB + nW + row) * K + 2 * half;
    const float* b1 = b0 + 16LL * K;
#pragma unroll 4
    for (int k = 0; k < K; k += 4) {
      v2f aa0 = *(const v2f*)(a0 + k);
      v2f aa1 = *(const v2f*)(a1 + k);
      v2f bb0 = *(const v2f*)(b0 + k);
      v2f bb1 = *(const v2f*)(b1 + k);
      acc00 = wmma4(aa0, bb0, acc00);
      acc01 = wmma4(aa0, bb1, acc01);
      acc10 = wmma4(aa1, bb0, acc10);
      acc11 = wmma4(aa1, bb1, acc11);
    }
  } else {
    // staging map: 16(K) x 128(N) panel, 256 threads x 2 float4, 2 buffers
    const int tid = wave * 32 + lane;            // 0..255
    const int r0  = (tid * 4) >> 7;              // 0..7
    const int c0  = (tid * 4) & 127;
    const float* gB = Bb + nB + c0;
    const uint32_t lA = (uint32_t)(uintptr_t)&bs[r0 * LDB + c0];
    const uint32_t lB = (uint32_t)(uintptr_t)&bs[(r0 + 8) * LDB + c0];
    const uint32_t bufBytes = (uint32_t)(KC * LDB * sizeof(float));

    // prologue: stage panel 0 into buffer 0
    async_b128(lA, gB + (long long)r0 * N);
    async_b128(lB, gB + (long long)(r0 + 8) * N);

    int buf = 0;
    for (int k = 0; k < K; k += KC) {
      asm volatile("s_wait_asynccnt 0x0" ::: "memory");  // current panel landed
      __syncthreads();                                   // visible to all waves
      if (k + KC < K) {                                  // prefetch next panel
        const uint32_t bo = (buf ^ 1) ? bufBytes : 0u;
        async_b128(lA + bo, gB + (long long)(k + KC + r0) * N);
        async_b128(lB + bo, gB + (long long)(k + KC + r0 + 8) * N);
      }
      const float* lsBase = &bs[buf * (KC * LDB) + nW + row];
#pragma unroll
      for (int kk = 0; kk < KC; kk += 4) {
        v2f aa0 = *(const v2f*)(a0 + k + kk);
        v2f aa1 = *(const v2f*)(a1 + k + kk);
        const float* ls = lsBase + (kk + 2 * half) * LDB;
        v2f bb0, bb1;
        bb0.x = ls[0];        bb0.y = ls[LDB];
        bb1.x = ls[16];       bb1.y = ls[LDB + 16];
        acc00 = wmma4(aa0, bb0, acc00);
        acc01 = wmma4(aa0, bb1, acc01);
        acc10 = wmma4(aa1, bb0, acc10);
        acc11 = wmma4(aa1, bb1, acc11);
      }
      __syncthreads();      // all waves done reading buf before it is refilled
      buf ^= 1;
    }
  }

  float* c0p = Cb + (long long)mW * N + nB + nW + row;
#pragma unroll
  for (int v = 0; v < 8; ++v) {
    long long rA = (long long)(v + 8 * half) * N;
    long long rB = (long long)(16 + v + 8 * half) * N;
    c0p[rA]      = acc00[v];
    c0p[rA + 16] = acc01[v];
    c0p[rB]      = acc10[v];
    c0p[rB + 16] = acc11[v];
  }
}

// ---------------------------------------------------------------------------
// Launcher
// ---------------------------------------------------------------------------
extern "C" void kernel_launch(void* const* d_in, const int* in_sizes, int n_in,
                              void* d_out, int out_size, void* d_ws, size_t ws_size,
                              hipStream_t stream) {
  const float* Z = (const float*)d_in[0];          // [16,4,512,512]
  float* out = (float*)d_out;                      // [16,4,1024,1024]

  const long long F  = (long long)NXF * NXF;       // 262144
  const long long FE = (long long)NEV * NXF;       // 524288
  const long long FO = (long long)NEV * NEV;       // 1048576

  float* w   = (float*)d_ws;
  float* t    = w;                      // 1024 (516 used)
  float* Bmat = t + 1024;               // 512*512
  float* Bt   = Bmat + F;               // 512*512
  float* Be   = Bt + F;                 // 1024*512
  float* Gb   = Be + FE;                // 512*4
  float* Lo   = Gb + 2048;              // 512*4
  float* S    = Lo + 2048;              // 64 * 512*512   (Bt * Z)
  float* T    = S + (long long)BC * F;  // 64 * 512*512   (T -> E -> Coef in place)
  float* U    = T + (long long)BC * F;  // 64 * 1024*512  (Be * Coef)

  dim3 wblk(32, 8, 1);                  // 8 wave32 waves

  // 1) knots + bases
  build_knots<<<2, 256, 0, stream>>>(t);
  build_basis<<<NXF, 128, 0, stream>>>(t, Bmat, NXF);
  build_basis<<<NEV, 128, 0, stream>>>(t, Be, NEV);
  transpose512<<<(int)(F / 256), 256, 0, stream>>>(Bmat, Bt);

  // 2) Gram + banded Cholesky (same for x and y axes: identical grids)
  gram_band<<<2, 256, 0, stream>>>(Bmat, Gb);
  chol_band<<<1, 1, 0, stream>>>(Gb, Lo);

  // 3) S = B^T * Z          [bc,512,512]
  gemm_wmma<0><<<dim3(NXF / 128, NXF / 64, BC), wblk, 0, stream>>>(
      Bt, 0LL, Z, F, S, F, NXF, NXF, NXF);

  // 4) T = S * B            [bc,512,512]
  gemm_wmma<0><<<dim3(NXF / 128, NXF / 64, BC), wblk, 0, stream>>>(
      S, F, Bmat, 0LL, T, F, NXF, NXF, NXF);

  // 5) solve (B^T B) E = T along axis 0, then axis 1 -> Coef (in place)
  band_solve_cols<<<(BC * NXF) / 256, 256, 0, stream>>>(T, Lo);
  band_solve_rows<<<(BC * NXF) / 256, 256, 0, stream>>>(T, Lo);

  // 6) U = Be * Coef        [bc,1024,512]
  gemm_wmma<0><<<dim3(NXF / 128, NEV / 64, BC), wblk, 0, stream>>>(
      Be, 0LL, T, F, U, FE, NEV, NXF, NXF);

  // 7) out = Be * U^T       [bc,1024(y),1024(x)]  (delivers the final transpose)
  gemm_wmma<1><<<dim3(NEV / 128, NEV / 64, BC), wblk, 0, stream>>>(
      Be, 0LL, U, FE, out, FO, NEV, NEV, NXF);
}